// LocallyConnected2d_63977832841365
// MI455X (gfx1250) — compile-verified
//
#include <hip/hip_runtime.h>
#include <hip/hip_bf16.h>

typedef __attribute__((ext_vector_type(2))) float v2f;
typedef __attribute__((ext_vector_type(8))) float v8f;

namespace {
constexpr int BATCH = 8;
constexpr int C_IN  = 32;
constexpr int H     = 48;
constexpr int W     = 48;
constexpr int C_OUT = 32;
constexpr int L     = H * W;              // 2304
constexpr int XB    = C_IN * H * W;       // per-batch x stride   = 73728 elems
constexpr int XC    = H * W;              // per-channel x stride = 2304 elems
constexpr int DSTR  = L * C_OUT;          // weight d-stride      = 73728 elems
constexpr int WAVES_PER_BLOCK = 4;        // 128 threads
}

__global__ __launch_bounds__(WAVES_PER_BLOCK * 32)
void lc2d_wmma_f32_kernel(const float* __restrict__ x,
                          const float* __restrict__ wgt,
                          const float* __restrict__ bias,
                          float* __restrict__ out)
{
    const int lane = threadIdx.x & 31;
    const int wave = blockIdx.x * WAVES_PER_BLOCK + (threadIdx.x >> 5);
    const int loc  = wave;                    // one spatial location per wave
    if (loc >= L) return;                     // uniform; never taken with exact grid

    const int h = loc / W;
    const int w = loc - h * W;

    const int mrow  = lane & 15;              // A: batch row | B: output-channel col
    const int mrowc = mrow < (BATCH - 1) ? mrow : (BATCH - 1);   // clamp for safe addr
    const int khalf = (lane >> 4) << 1;       // K slots {0,1} or {2,3}

    // ---- Precompute per-tap (r = kh*3+kw) lane offsets and 0/1 masks (branchless) ----
    int   aoff[9];
    float amsk[9];
#pragma unroll
    for (int r = 0; r < 9; ++r) {
        const int kh = r / 3, kw = r - kh * 3;          // compile-time after unroll
        const int hh = h + kh - 1, ww = w + kw - 1;     // wave-uniform
        const bool inb = ((unsigned)hh < (unsigned)H) & ((unsigned)ww < (unsigned)W);
        const int hhc = hh < 0 ? 0 : (hh > H - 1 ? H - 1 : hh);
        const int wwc = ww < 0 ? 0 : (ww > W - 1 ? W - 1 : ww);
        aoff[r] = mrowc * XB + khalf * XC + hhc * W + wwc;
        amsk[r] = (inb && (mrow < BATCH)) ? 1.0f : 0.0f;
    }

    v8f acc0 = {};                            // o = 0..15
    v8f acc1 = {};                            // o = 16..31

    // B base: folds K-slot half, location column, and lane's output channel.
    const float* __restrict__ bbase =
        wgt + (size_t)khalf * 9 * DSTR + (size_t)loc * C_OUT + mrow;

    // Reduction order: d = c*9 + r, chunks of 4 consecutive channels per WMMA step.
    // K slot k of chunk (cb, r) carries d = (cb*4 + k)*9 + r for BOTH A and B.
#pragma unroll
    for (int cb = 0; cb < C_IN / 4; ++cb) {               // 8 outer steps
        const float* __restrict__ ap = x + cb * 4 * XC;   // + aoff[r] per lane
        const float* __restrict__ bp = bbase + (size_t)cb * 36 * DSTR;
#pragma unroll
        for (int r = 0; r < 9; ++r) {                     // 9 taps, imm offsets only
            v2f a, b0, b1;
            a[0] = ap[aoff[r]]      * amsk[r];            // c = cb*4 + khalf + 0
            a[1] = ap[aoff[r] + XC] * amsk[r];            // c = cb*4 + khalf + 1

            b0[0] = __builtin_nontemporal_load(bp + (size_t)r * DSTR);
            b0[1] = __builtin_nontemporal_load(bp + (size_t)(r + 9) * DSTR);
            b1[0] = __builtin_nontemporal_load(bp + (size_t)r * DSTR + 16);
            b1[1] = __builtin_nontemporal_load(bp + (size_t)(r + 9) * DSTR + 16);

            acc0 = __builtin_amdgcn_wmma_f32_16x16x4_f32(
                       false, a, false, b0, (short)0, acc0, false, false);
            acc1 = __builtin_amdgcn_wmma_f32_16x16x4_f32(
                       false, a, false, b1, (short)0, acc1, false, false);
        }
    }

    // ---- Writeback: C/D layout: VGPR i -> M = i + 8*(lane>=16), N = lane&15 ----
    const int mbase = (lane >> 4) << 3;       // 0 or 8
    const int n     = lane & 15;
    const float bi0 = bias[(size_t)n * L + loc];
    const float bi1 = bias[(size_t)(n + 16) * L + loc];
#pragma unroll
    for (int i = 0; i < 8; ++i) {
        const int m = mbase + i;              // batch index (8..15 are padding rows)
        if (m < BATCH) {
            out[((size_t)m * C_OUT + n)      * L + loc] = acc0[i] + bi0;
            out[((size_t)m * C_OUT + n + 16) * L + loc] = acc1[i] + bi1;
        }
    }
}

extern "C" void kernel_launch(void* const* d_in, const int* in_sizes, int n_in,
                              void* d_out, int out_size, void* d_ws, size_t ws_size,
                              hipStream_t stream) {
    (void)in_sizes; (void)n_in; (void)out_size; (void)d_ws; (void)ws_size;
    const float* x    = (const float*)d_in[0];
    const float* wgt  = (const float*)d_in[1];
    const float* bias = (const float*)d_in[2];
    float* out        = (float*)d_out;

    const int total_waves = L;                                                 // 2304
    const int blocks = (total_waves + WAVES_PER_BLOCK - 1) / WAVES_PER_BLOCK;  // 576
    lc2d_wmma_f32_kernel<<<blocks, WAVES_PER_BLOCK * 32, 0, stream>>>(x, wgt, bias, out);
}